// AdjacencyProcessing_43576738185343
// MI455X (gfx1250) — compile-verified
//
#include <hip/hip_runtime.h>

// Problem constants (reference: N=8192 fp32 adjacency)
#define GN 8192
#define REGF 0.001f
#define SEGS 8          // K-split for row-sum pass (parallelism)
#define SEGLEN (GN / SEGS)       // 1024 columns per segment
#define QLEN (SEGLEN / 4)        // 256 columns per WMMA K-slot

typedef float v2f __attribute__((ext_vector_type(2)));
typedef float v4f __attribute__((ext_vector_type(4)));
typedef float v8f __attribute__((ext_vector_type(8)));

// ---------------------------------------------------------------------------
// Pass 1: partial row sums. One wave32 handles a 16-row tile over one
// 1024-column segment. Lane L (m = L&15, h = L>>4) streams two disjoint
// 256-element quarters of row (tile*16 + m) into acc.x / acc.y, which are
// exactly the K=2h and K=2h+1 slots of the V_WMMA_F32_16X16X4_F32 A-matrix
// layout. A single WMMA with B == ones then produces per-row sums in the
// 16x16 f32 D layout (value replicated across N). Lanes 0 and 16 store the
// 16 row sums.
// ---------------------------------------------------------------------------
__global__ void __launch_bounds__(256)
rowsum_partial_kernel(const float* __restrict__ A, float* __restrict__ partial) {
    const unsigned lane   = threadIdx.x & 31u;
    const unsigned waveId = blockIdx.x * 8u + (threadIdx.x >> 5);   // 0..4095
    const unsigned tile   = waveId & (GN / 16 - 1);                 // 0..511
    const unsigned seg    = waveId >> 9;                            // 0..7

    const unsigned m = lane & 15u;
    const unsigned h = lane >> 4;          // 0 or 1
    const unsigned row = tile * 16u + m;

    // Lane's two streaming chunks: quarters (2h) and (2h+1) of this segment.
    const float* base = A + (size_t)row * GN + (size_t)seg * SEGLEN + (size_t)h * (2 * QLEN);
    const v4f* p0 = (const v4f*)base;            // quarter 2h   (256 floats)
    const v4f* p1 = p0 + (QLEN / 4);             // quarter 2h+1 (256 floats)

    float ax = 0.0f, ay = 0.0f;
#pragma unroll 4
    for (int t = 0; t < QLEN / 4; ++t) {
        v4f a0 = p0[t];
        v4f a1 = p1[t];
        ax += (a0.x + a0.y) + (a0.z + a0.w);
        ay += (a1.x + a1.y) + (a1.z + a1.w);
    }

    // A-matrix (16x4 f32): lane holds K = {2h, 2h+1} for row m.
    v2f amat; amat.x = ax; amat.y = ay;
    // B-matrix (4x16) all ones -> D[m][n] = sum_k A[m][k] for every n.
    v2f bmat; bmat.x = 1.0f; bmat.y = 1.0f;
    v8f cmat = {};

    v8f d = __builtin_amdgcn_wmma_f32_16x16x4_f32(
        /*neg_a=*/false, amat, /*neg_b=*/false, bmat,
        /*c_mod=*/(short)0, cmat, /*reuse_a=*/false, /*reuse_b=*/false);

    // D layout: VGPR j -> row j (lanes 0-15) / row j+8 (lanes 16-31),
    // replicated across the 16 columns. Two lanes write 8 sums each.
    v4f lo; lo.x = d[0]; lo.y = d[1]; lo.z = d[2]; lo.w = d[3];
    v4f hi; hi.x = d[4]; hi.y = d[5]; hi.z = d[6]; hi.w = d[7];

    float* pr = partial + (size_t)seg * GN + (size_t)tile * 16u;
    if (lane == 0u) {
        *(v4f*)(pr)     = lo;
        *(v4f*)(pr + 4) = hi;
    } else if (lane == 16u) {
        *(v4f*)(pr + 8)  = lo;
        *(v4f*)(pr + 12) = hi;
    }
}

// ---------------------------------------------------------------------------
// Pass 1.5: reduce the SEGS partials per row and turn them into the two
// per-row scalars:  scaleA = inv*norm,  scaleD = inv*(1+REG).
// ---------------------------------------------------------------------------
__global__ void __launch_bounds__(256)
finalize_scales_kernel(const float* __restrict__ partial,
                       float* __restrict__ scaleA, float* __restrict__ scaleD) {
    const int i = blockIdx.x * 256 + threadIdx.x;
    if (i >= GN) return;
    float rs = 0.0f;
#pragma unroll
    for (int s = 0; s < SEGS; ++s) rs += partial[(size_t)s * GN + i];
    const float norm  = 1.0f / fmaxf(1.0f, rs);
    const float denom = rs + 1.0f;
    const float inv   = (denom == 0.0f) ? 0.0f : 1.0f / denom;
    scaleA[i] = inv * norm;
    scaleD[i] = inv * (1.0f + REGF);
}

// ---------------------------------------------------------------------------
// Pass 2: out[i][j] = scaleA[i]*A[i][j] (+ scaleD[i] on the diagonal).
// Each block owns half a row (row is blockIdx-uniform -> scalar loads of the
// scales). float4 loads; non-temporal float4 stores keep the 256 MB output
// stream out of L2.
// ---------------------------------------------------------------------------
__global__ void __launch_bounds__(256)
apply_kernel(const float* __restrict__ A,
             const float* __restrict__ scaleA, const float* __restrict__ scaleD,
             float* __restrict__ out) {
    const unsigned row   = blockIdx.x >> 1;
    const unsigned jBase = (blockIdx.x & 1u) * (GN / 2) + threadIdx.x * 16u;

    const float sA = scaleA[row];
    const float sD = scaleD[row];

    const float* src = A   + (size_t)row * GN + jBase;
    float*       dst = out + (size_t)row * GN + jBase;

#pragma unroll
    for (int g = 0; g < 4; ++g) {
        v4f v = *(const v4f*)(src + 4 * g);
        v.x *= sA; v.y *= sA; v.z *= sA; v.w *= sA;
        const int dcol = (int)row - (int)(jBase + 4u * g);
        if (dcol >= 0 && dcol < 4) v[dcol] += sD;
        __builtin_nontemporal_store(v, (v4f*)(dst + 4 * g));
    }
}

extern "C" void kernel_launch(void* const* d_in, const int* in_sizes, int n_in,
                              void* d_out, int out_size, void* d_ws, size_t ws_size,
                              hipStream_t stream) {
    (void)in_sizes; (void)n_in; (void)out_size; (void)ws_size;
    const float* A = (const float*)d_in[0];
    float* out = (float*)d_out;

    // Workspace layout: partial[SEGS][GN] | scaleA[GN] | scaleD[GN]  (~320 KB)
    float* partial = (float*)d_ws;
    float* scaleA  = partial + (size_t)SEGS * GN;
    float* scaleD  = scaleA + GN;

    // Pass 1: 512 row-tiles * 8 segments = 4096 waves = 512 blocks of 256.
    rowsum_partial_kernel<<<(GN / 16) * SEGS / 8, 256, 0, stream>>>(A, partial);
    // Pass 1.5: one thread per row.
    finalize_scales_kernel<<<GN / 256, 256, 0, stream>>>(partial, scaleA, scaleD);
    // Pass 2: 2 blocks per row, 16 floats per thread.
    apply_kernel<<<GN * 2, 256, 0, stream>>>(A, scaleA, scaleD, out);
}